// CLOPLayer_14096082666280
// MI455X (gfx1250) — compile-verified
//
#include <hip/hip_runtime.h>
#include <stdint.h>

// ---------------- problem constants (match reference) ----------------
#define H_      224
#define W_      224
#define N_      (H_ * W_)          // 50176 grid cells
#define PLANES  (128 * 3)          // b*c = 384 independent planes
#define TILE    1024               // idx entries staged per block (4 KB LDS)
#define PB      16                 // planes processed per block
// cumulative probs for p=0.3: [0.7, 0.775, 0.85, 0.925, 1.0]

// ---------------- threefry2x32 (20 rounds, canonical schedule) ----------------
__device__ __forceinline__ void tf2x32(uint32_t k0, uint32_t k1,
                                       uint32_t c0, uint32_t c1,
                                       uint32_t* o0, uint32_t* o1) {
  uint32_t ks2 = 0x1BD11BDAu ^ k0 ^ k1;
  uint32_t x0 = c0 + k0, x1 = c1 + k1;
#define TF_RND(r) { x0 += x1; x1 = (x1 << (r)) | (x1 >> (32 - (r))); x1 ^= x0; }
  TF_RND(13) TF_RND(15) TF_RND(26) TF_RND(6)  x0 += k1;  x1 += ks2 + 1u;
  TF_RND(17) TF_RND(29) TF_RND(16) TF_RND(24) x0 += ks2; x1 += k0  + 2u;
  TF_RND(13) TF_RND(15) TF_RND(26) TF_RND(6)  x0 += k0;  x1 += k1  + 3u;
  TF_RND(17) TF_RND(29) TF_RND(16) TF_RND(24) x0 += k1;  x1 += ks2 + 4u;
  TF_RND(13) TF_RND(15) TF_RND(26) TF_RND(6)  x0 += ks2; x1 += k0  + 5u;
#undef TF_RND
  *o0 = x0; *o1 = x1;
}

// JAX key(42) split -> k1 (order), k2 (move types)
__device__ __forceinline__ void clop_keys(uint32_t* k1a, uint32_t* k1b,
                                          uint32_t* k2a, uint32_t* k2b) {
  uint32_t a0, b0, a1, b1;
  tf2x32(0u, 42u, 0u, 2u, &a0, &b0);
  tf2x32(0u, 42u, 1u, 3u, &a1, &b1);
  *k1a = a0; *k1b = a1;   // first subkey
  *k2a = b0; *k2b = b1;   // second subkey
}

// ---------------- CDNA5 async global->LDS helpers ----------------
typedef int v4i __attribute__((vector_size(4 * sizeof(int))));
typedef __attribute__((address_space(1))) v4i as1_v4i;   // global (__device__) int4
typedef __attribute__((address_space(3))) v4i as3_v4i;   // LDS (__shared__) int4

__device__ __forceinline__ void async_g2l_b128(const int* gsrc, int* ldst,
                                               uint32_t lds_byte_off) {
#if defined(__has_builtin) && __has_builtin(__builtin_amdgcn_global_load_async_to_lds_b128)
  __builtin_amdgcn_global_load_async_to_lds_b128(
      (as1_v4i*)gsrc, (as3_v4i*)ldst, 0, 0);
  (void)lds_byte_off;
#else
  // VDST = VGPR holding LDS byte address, VADDR = 64-bit global address pair.
  asm volatile("global_load_async_to_lds_b128 %0, %1, off"
               :: "v"(lds_byte_off), "v"(gsrc) : "memory");
#endif
}

__device__ __forceinline__ void wait_asynccnt0() {
#if defined(__has_builtin) && __has_builtin(__builtin_amdgcn_s_wait_asynccnt)
  __builtin_amdgcn_s_wait_asynccnt(0);
#else
  asm volatile("s_wait_asynccnt 0" ::: "memory");
#endif
}

// ---------------- kernel 1: visitation order (Fisher-Yates in LDS) ----------------
__global__ void clop_make_order(int* __restrict__ ord_g) {
  extern __shared__ int lord[];                    // N_ ints = 196 KB (< 320 KB WGP LDS)
  for (int i = threadIdx.x; i < N_; i += blockDim.x) lord[i] = i;
  __syncthreads();
  if (threadIdx.x == 0) {
    uint32_t k1a, k1b, k2a, k2b;
    clop_keys(&k1a, &k1b, &k2a, &k2b);
    for (int i = N_ - 1; i > 0; --i) {             // sequential by construction
      uint32_t r0, r1;
      tf2x32(k1a, k1b, (uint32_t)i, 0x9E3779B9u, &r0, &r1);
      int j = (int)(r0 % (uint32_t)(i + 1));
      int t = lord[i]; lord[i] = lord[j]; lord[j] = t;
    }
  }
  __syncthreads();
  for (int i = threadIdx.x; i < N_; i += blockDim.x) ord_g[i] = lord[i];
}

// ---------------- kernel 2: sequential neighbor-swap scan (indexes in LDS) ----------------
__global__ void clop_scan(const int* __restrict__ ord_g, int* __restrict__ idx_g) {
  extern __shared__ int lidx[];                    // N_ ints
  for (int i = threadIdx.x; i < N_; i += blockDim.x) lidx[i] = i;
  __syncthreads();
  if (threadIdx.x == 0) {
    uint32_t k1a, k1b, k2a, k2b;
    clop_keys(&k1a, &k1b, &k2a, &k2b);
    const int half = N_ / 2;
    for (int t = 0; t < N_; ++t) {
      int idx = ord_g[t];                          // streaming read of visitation order
      // JAX-style random bits for element t of an (N_,) draw under k2
      uint32_t lo, hi, bits;
      if (t < half) { tf2x32(k2a, k2b, (uint32_t)t, (uint32_t)(t + half), &lo, &hi); bits = lo; }
      else          { tf2x32(k2a, k2b, (uint32_t)(t - half), (uint32_t)t, &lo, &hi); bits = hi; }
      float u = __uint_as_float((bits >> 9) | 0x3F800000u) - 1.0f;   // uniform [0,1)
      float r = 1.0f - u;                          // p_cuml[-1] * (1 - u)
      int mv = (r <= 0.700f) ? 0 :
               (r <= 0.775f) ? 1 :
               (r <= 0.850f) ? 2 :
               (r <= 0.925f) ? 3 : 4;              // searchsorted on cumprobs
      if (mv != 0) {
        int i0 = idx / W_, j0 = idx % W_;
        int ip = (mv == 1) ? (i0 + 1) % H_ : (mv == 2) ? (i0 + H_ - 1) % H_ : i0;
        int jp = (mv == 3) ? (j0 + 1) % W_ : (mv == 4) ? (j0 + W_ - 1) % W_ : j0;
        int idx2 = ip * W_ + jp;
        int a = lidx[idx]; lidx[idx] = lidx[idx2]; lidx[idx2] = a;
      }
    }
  }
  __syncthreads();
  for (int i = threadIdx.x; i < N_; i += blockDim.x) idx_g[i] = lidx[i];
}

// ---------------- kernel 3: bandwidth gather (the 154 MB part) ----------------
__global__ __launch_bounds__(256) void clop_gather(const float* __restrict__ x,
                                                   const int* __restrict__ idx_g,
                                                   float* __restrict__ out) {
  __shared__ int s_idx[TILE];                      // 4 KB idx tile, reused for PB planes
  const int tid   = threadIdx.x;
  const int tbase = blockIdx.x * TILE;

  // Stage idx tile into LDS via CDNA5 async copy: 256 lanes x b128 = TILE ints.
  async_g2l_b128(idx_g + tbase + tid * 4, &s_idx[tid * 4], (uint32_t)(tid * 16));
  wait_asynccnt0();
  __syncthreads();

  const int4 v = *(const int4*)&s_idx[tid * 4];    // ds_load_b128, stays in registers
  const int p0 = blockIdx.y * PB;
#pragma unroll 4
  for (int p = p0; p < p0 + PB; ++p) {
    const float* xp = x + (size_t)p * N_;
    float4 o;
    o.x = xp[v.x];                                 // scattered-but-local b32 gathers
    o.y = xp[v.y];
    o.z = xp[v.z];
    o.w = xp[v.w];
    *(float4*)&out[(size_t)p * N_ + tbase + tid * 4] = o;   // coalesced b128 store
  }
}

// ---------------- launch ----------------
extern "C" void kernel_launch(void* const* d_in, const int* in_sizes, int n_in,
                              void* d_out, int out_size, void* d_ws, size_t ws_size,
                              hipStream_t stream) {
  (void)in_sizes; (void)n_in; (void)out_size; (void)ws_size;
  const float* x  = (const float*)d_in[0];         // [128,3,224,224] f32
  float* out      = (float*)d_out;
  int*   ord      = (int*)d_ws;                    // N_ ints
  int*   idx      = ord + N_;                      // N_ ints

  const size_t ldsBytes = (size_t)N_ * sizeof(int);   // 200704 B dynamic LDS

  clop_make_order<<<1, 256, ldsBytes, stream>>>(ord);
  clop_scan<<<1, 256, ldsBytes, stream>>>(ord, idx);

  dim3 grid(N_ / TILE, PLANES / PB);               // (49, 24) = 1176 blocks
  clop_gather<<<grid, 256, 0, stream>>>(x, idx, out);
}